// GNN_72705206387210
// MI455X (gfx1250) — compile-verified
//
#include <hip/hip_runtime.h>

#define GN 40000
#define GE 640000
#define GH 128
#define GETOT (GE + GN)
#define GNF (GN * GH)

typedef __attribute__((ext_vector_type(2))) float v2f;
typedef __attribute__((ext_vector_type(8))) float v8f;

// ---- monotone float<->int encoding for atomicMax-based segment max ----
__device__ __forceinline__ int enc_f32(float x) {
  int i = __float_as_int(x);
  return i >= 0 ? i : (i ^ 0x7fffffff);
}
__device__ __forceinline__ float dec_f32(int i) {
  return __int_as_float(i >= 0 ? i : (i ^ 0x7fffffff));
}

// ---------------- utility fills ----------------
__global__ void gnn_fill_f32(float* __restrict__ p, float v, int n) {
  int i = blockIdx.x * blockDim.x + threadIdx.x;
  if (i < n) p[i] = v;
}
__global__ void gnn_fill_i32(int* __restrict__ p, int v, int n) {
  int i = blockIdx.x * blockDim.x + threadIdx.x;
  if (i < n) p[i] = v;
}

// ---------------- degree / dinv ----------------
__global__ void gnn_deg(const long long* __restrict__ row, float* __restrict__ deg, int E) {
  int e = blockIdx.x * blockDim.x + threadIdx.x;
  if (e < E) atomicAdd(&deg[(int)row[e]], 1.0f);
}
__global__ void gnn_rsqrt(float* __restrict__ p, int n) {
  int i = blockIdx.x * blockDim.x + threadIdx.x;
  if (i < n) p[i] = rsqrtf(fmaxf(p[i], 1.0f));
}

// ---------------- fold [g,g]@W(256x128) -> g@(W0+W1) ----------------
__global__ void gnn_wsum(const float* __restrict__ W, float* __restrict__ Ws) {
  int i = blockIdx.x * blockDim.x + threadIdx.x;
  if (i < GH * GH) Ws[i] = W[i] + W[i + GH * GH];
}

// ---------------- f32 WMMA GEMM: C[M x 128] = A[M x K] @ W[K x 128] (+bias) ----
// grid.x = M/16, block = 256 (8 waves). Wave w owns output cols [16w,16w+16).
__global__ __launch_bounds__(256)
void gnn_gemm_wmma(const float* __restrict__ A, const float* __restrict__ W,
                   const float* __restrict__ bias, float* __restrict__ C, int K) {
  __shared__ float As[16 * (128 + 4)];                 // K <= 128, +4 pad: conflict-free
  const int tid = threadIdx.x;
  const int m0  = blockIdx.x * 16;
  const int ldp = K + 4;
  const float* Ablk = A + (size_t)m0 * K;
  for (int i = tid; i < 16 * K; i += 256) {
    int r = i / K, k = i - r * K;
    As[r * ldp + k] = Ablk[i];
  }
  __syncthreads();

  const int wave = tid >> 5;
  const int lane = tid & 31;
  const int c    = lane & 15;          // row-in-tile (A) / col-in-tile (B,C)
  const int hi   = lane >> 4;          // K-half selector per 16x16x4 layout
  const int n    = wave * 16 + c;      // output column

  v8f acc = {};
  for (int k = 0; k < K; k += 4) {
    const int kk = k + 2 * hi;
    v2f av, bv;
    av.x = As[c * ldp + kk];
    av.y = As[c * ldp + kk + 1];
    bv.x = W[(size_t)kk * GH + n];
    bv.y = W[(size_t)(kk + 1) * GH + n];
    acc = __builtin_amdgcn_wmma_f32_16x16x4_f32(false, av, false, bv,
                                                (short)0, acc, false, false);
  }
  const float b = bias ? bias[n] : 0.0f;
#pragma unroll
  for (int r = 0; r < 8; ++r) {
    C[(size_t)(m0 + r + 8 * hi) * GH + n] = acc[r] + b;
  }
}

// ---------------- GCN edge scatter: acc[r] += xw[c] * dinv[r]*dinv[c] ----------
__global__ void gnn_gcn_scatter(const float* __restrict__ xw,
                                const long long* __restrict__ row,
                                const long long* __restrict__ col,
                                const float* __restrict__ dinv,
                                float* __restrict__ acc, int E, int Etot) {
  int wid  = (blockIdx.x * blockDim.x + threadIdx.x) >> 5;
  int lane = threadIdx.x & 31;
  if (wid >= Etot) return;
  int r, c;
  if (wid < E) { r = (int)row[wid]; c = (int)col[wid]; }
  else         { r = c = wid - E; }
  const float w = dinv[r] * dinv[c];
  const float4 v = ((const float4*)(xw + (size_t)c * GH))[lane];
  float* dst = acc + (size_t)r * GH + lane * 4;
  atomicAdd(dst + 0, v.x * w);
  atomicAdd(dst + 1, v.y * w);
  atomicAdd(dst + 2, v.z * w);
  atomicAdd(dst + 3, v.w * w);
}

__global__ void gnn_gcn_finalize(float* __restrict__ p, const float* __restrict__ b, int n) {
  int i = blockIdx.x * blockDim.x + threadIdx.x;
  if (i < n) p[i] = fmaxf(p[i] + b[i & (GH - 1)], 0.0f);
}

// ---------------- GAT pass 1: alpha = leaky(dot(h[r],h[c])), segment max -------
__global__ void gnn_gat_alpha(const float* __restrict__ h,
                              const long long* __restrict__ row,
                              const long long* __restrict__ col,
                              float* __restrict__ alpha, int* __restrict__ amax,
                              int E, int Etot) {
  int wid  = (blockIdx.x * blockDim.x + threadIdx.x) >> 5;
  int lane = threadIdx.x & 31;
  if (wid >= Etot) return;
  int r, c;
  if (wid < E) {
    r = (int)row[wid]; c = (int)col[wid];
    if (r == c) {                                    // removed self-loop: masked
      if (lane == 0) alpha[wid] = __int_as_float(0xff800000);
      return;
    }
  } else {
    r = c = wid - E;                                 // appended self-loop: valid
  }
  const float4 a = ((const float4*)(h + (size_t)r * GH))[lane];
  const float4 b = ((const float4*)(h + (size_t)c * GH))[lane];
  float d = a.x * b.x + a.y * b.y + a.z * b.z + a.w * b.w;
  for (int m = 16; m; m >>= 1) d += __shfl_xor(d, m, 32);
  if (lane == 0) {
    float al = d > 0.0f ? d : 0.2f * d;              // leaky_relu(0.2)
    alpha[wid] = al;
    atomicMax(&amax[r], enc_f32(al));
  }
}

// ---------------- GAT pass 2: a = exp(alpha - amax[r]); denom[r] += a ----------
__global__ void gnn_gat_exp(float* __restrict__ alpha,
                            const long long* __restrict__ row,
                            const int* __restrict__ amax,
                            float* __restrict__ denom, int E, int Etot) {
  int e = blockIdx.x * blockDim.x + threadIdx.x;
  if (e >= Etot) return;
  const float al = alpha[e];
  const int r = (e < E) ? (int)row[e] : (e - E);
  float a = 0.0f;
  if (al != __int_as_float(0xff800000)) a = expf(al - dec_f32(amax[r]));
  alpha[e] = a;
  if (a != 0.0f) atomicAdd(&denom[r], a);
}

// ---------------- GAT pass 3: out[r] += h[c] * a/denom[r] ----------------------
__global__ void gnn_gat_scatter(const float* __restrict__ h,
                                const long long* __restrict__ row,
                                const long long* __restrict__ col,
                                const float* __restrict__ a_e,
                                const float* __restrict__ denom,
                                float* __restrict__ out, int E, int Etot) {
  int wid  = (blockIdx.x * blockDim.x + threadIdx.x) >> 5;
  int lane = threadIdx.x & 31;
  if (wid >= Etot) return;
  const float a = a_e[wid];
  if (a == 0.0f) return;
  int r, c;
  if (wid < E) { r = (int)row[wid]; c = (int)col[wid]; }
  else         { r = c = wid - E; }
  const float w = a / fmaxf(denom[r], 1e-16f);
  const float4 v = ((const float4*)(h + (size_t)c * GH))[lane];
  float* dst = out + (size_t)r * GH + lane * 4;
  atomicAdd(dst + 0, v.x * w);
  atomicAdd(dst + 1, v.y * w);
  atomicAdd(dst + 2, v.z * w);
  atomicAdd(dst + 3, v.w * w);
}

// ---------------- column mean of final features, duplicated [h,h] --------------
__global__ void gnn_colmean(const float* __restrict__ g, float* __restrict__ out, int n) {
  __shared__ float sm[256];
  const int col = blockIdx.x;
  float s = 0.0f;
  for (int r = threadIdx.x; r < n; r += blockDim.x) s += g[(size_t)r * GH + col];
  sm[threadIdx.x] = s;
  __syncthreads();
  for (int off = 128; off; off >>= 1) {
    if ((int)threadIdx.x < off) sm[threadIdx.x] += sm[threadIdx.x + off];
    __syncthreads();
  }
  if (threadIdx.x == 0) {
    float m = sm[0] / (float)n;
    out[col] = m;
    out[col + GH] = m;
  }
}

extern "C" void kernel_launch(void* const* d_in, const int* in_sizes, int n_in,
                              void* d_out, int out_size, void* d_ws, size_t ws_size,
                              hipStream_t stream) {
  (void)in_sizes; (void)n_in; (void)out_size; (void)ws_size;

  const float*      x   = (const float*)d_in[0];
  const long long*  ei  = (const long long*)d_in[1];   // int64 edge_index [2, E]
  const long long*  row = ei;
  const long long*  col = ei + GE;
  const float* Wg[3] = {(const float*)d_in[2],  (const float*)d_in[6],  (const float*)d_in[10]};
  const float* bg[3] = {(const float*)d_in[3],  (const float*)d_in[7],  (const float*)d_in[11]};
  const float* Wa[3] = {(const float*)d_in[4],  (const float*)d_in[8],  (const float*)d_in[12]};
  const float* ba[3] = {(const float*)d_in[5],  (const float*)d_in[9],  (const float*)d_in[13]};

  char* ws = (char*)d_ws;                              // ~64.7 MB total
  float* B0    = (float*)(ws + 0);                     // 20.48 MB  (GAT out / next input)
  float* B1    = (float*)(ws + 20480000);              // 20.48 MB  (XW, then GAT h)
  float* B2    = (float*)(ws + 40960000);              // 20.48 MB  (GCN acc/out)
  float* alpha = (float*)(ws + 61440000);              // 2.72 MB   (per-edge alpha/a)
  float* dinv  = (float*)(ws + 64160000);              // 160 KB
  int*   amax  = (int*)  (ws + 64320000);              // 160 KB
  float* denom = (float*)(ws + 64480000);              // 160 KB
  float* Wsum  = (float*)(ws + 64640000);              // 64 KB

  const dim3 blk(256);
  const dim3 gridN((GN + 255) / 256);
  const dim3 gridE((GE + 255) / 256);
  const dim3 gridNF((GNF + 255) / 256);
  const dim3 gridEt((GETOT + 255) / 256);
  const dim3 gridEW((GETOT + 7) / 8);                  // one wave32 per edge
  const dim3 gridGemm(GN / 16);

  // degrees (self-loop included) and D^{-1/2}
  gnn_fill_f32<<<gridN, blk, 0, stream>>>(dinv, 1.0f, GN);
  gnn_deg<<<gridE, blk, 0, stream>>>(row, dinv, GE);
  gnn_rsqrt<<<gridN, blk, 0, stream>>>(dinv, GN);

  const float* P = x;
  for (int l = 0; l < 3; ++l) {
    const float* Wc = Wg[l];
    if (l > 0) {                                       // fold concat into weights
      gnn_wsum<<<dim3((GH * GH + 255) / 256), blk, 0, stream>>>(Wg[l], Wsum);
      Wc = Wsum;
    }
    // ---- GCN ----
    gnn_gemm_wmma<<<gridGemm, blk, 0, stream>>>(P, Wc, nullptr, B1, GH);
    gnn_fill_f32<<<gridNF, blk, 0, stream>>>(B2, 0.0f, GNF);
    gnn_gcn_scatter<<<gridEW, blk, 0, stream>>>(B1, row, col, dinv, B2, GE, GETOT);
    gnn_gcn_finalize<<<gridNF, blk, 0, stream>>>(B2, bg[l], GNF);
    // ---- GAT ----
    gnn_gemm_wmma<<<gridGemm, blk, 0, stream>>>(B2, Wa[l], ba[l], B1, GH);
    gnn_fill_i32<<<gridN, blk, 0, stream>>>(amax, (int)0x807FFFFF, GN);  // enc(-inf)
    gnn_fill_f32<<<gridN, blk, 0, stream>>>(denom, 0.0f, GN);
    gnn_gat_alpha<<<gridEW, blk, 0, stream>>>(B1, row, col, alpha, amax, GE, GETOT);
    gnn_gat_exp<<<gridEt, blk, 0, stream>>>(alpha, row, amax, denom, GE, GETOT);
    gnn_fill_f32<<<gridNF, blk, 0, stream>>>(B0, 0.0f, GNF);
    gnn_gat_scatter<<<gridEW, blk, 0, stream>>>(B1, row, col, alpha, denom, B0, GE, GETOT);
    P = B0;
  }
  gnn_colmean<<<dim3(GH), blk, 0, stream>>>(B0, (float*)d_out, GN);
}